// WKnn_53154515255414
// MI455X (gfx1250) — compile-verified
//
#include <hip/hip_runtime.h>
#include <math.h>

#define DF   12
#define NB   1024
#define SSZ  4095
#define KNB  32
#define NCLS 10

typedef __attribute__((ext_vector_type(2))) float v2f;
typedef __attribute__((ext_vector_type(8))) float v8f;

// ---------------------------------------------------------------------------
// Kernel A: h1 = relu(X @ W1^T + b1)   [1024,12] x [12,32] -> [1024,32]
// One wave (32 lanes) per 16x16 output tile, V_WMMA_F32_16X16X4_F32, K = 3x4.
// A layout (16x4 f32): VGPR0 lanes0-15 K=0 / lanes16-31 K=2; VGPR1 K=1 / K=3.
// B layout (4x16 f32): row K striped across lanes within a VGPR (mirror of A).
// C/D layout (16x16 f32): VGPR v: lanes0-15 -> M=v, lanes16-31 -> M=v+8.
// ---------------------------------------------------------------------------
__global__ __launch_bounds__(32) void k_gemm1_wmma(const float* __restrict__ X,
                                                   const float* __restrict__ W1,
                                                   const float* __restrict__ b1,
                                                   float* __restrict__ h1) {
  const int lane  = threadIdx.x;       // 0..31, exactly one wave32 per block
  const int tileM = blockIdx.x >> 1;   // 64 row tiles
  const int tileN = blockIdx.x & 1;    // 2 col tiles
  const int half  = lane >> 4;         // lane group 0/1
  const int l15   = lane & 15;
  const int row   = tileM * 16 + l15;  // A-matrix row handled by this lane
  const int col   = tileN * 16 + l15;  // B/C/D column handled by this lane

  v8f acc = {};
#pragma unroll
  for (int k0 = 0; k0 < DF; k0 += 4) {
    v2f a, b;
    a.x = X[row * DF + k0 + 2 * half];
    a.y = X[row * DF + k0 + 2 * half + 1];
    // B[k][n] = W1[n][k]  (h = X @ W1^T)
    b.x = W1[col * DF + k0 + 2 * half];
    b.y = W1[col * DF + k0 + 2 * half + 1];
    acc = __builtin_amdgcn_wmma_f32_16x16x4_f32(false, a, false, b,
                                                (short)0, acc, false, false);
  }
  const float bias = b1[col];
#pragma unroll
  for (int v = 0; v < 8; ++v) {
    const int r = tileM * 16 + v + 8 * half;
    const float val = acc[v] + bias;
    h1[r * 32 + col] = val > 0.f ? val : 0.f;
  }
}

// ---------------------------------------------------------------------------
// Kernel B: per-row  h2 = relu(h1@W2^T+b2); LN; h3 = LN(h2)@W3^T + b3
// ---------------------------------------------------------------------------
__global__ __launch_bounds__(256) void k_encoder_rest(
    const float* __restrict__ h1, const float* __restrict__ W2,
    const float* __restrict__ b2, const float* __restrict__ lng,
    const float* __restrict__ lnb, const float* __restrict__ W3,
    const float* __restrict__ b3, float* __restrict__ h3) {
  const int r = blockIdx.x * blockDim.x + threadIdx.x;
  if (r >= NB) return;
  float a[32];
#pragma unroll
  for (int k = 0; k < 32; ++k) a[k] = h1[r * 32 + k];
  float h2[16];
#pragma unroll
  for (int j = 0; j < 16; ++j) {
    float s = b2[j];
#pragma unroll
    for (int k = 0; k < 32; ++k) s += a[k] * W2[j * 32 + k];
    h2[j] = s > 0.f ? s : 0.f;
  }
  float mu = 0.f;
#pragma unroll
  for (int j = 0; j < 16; ++j) mu += h2[j];
  mu *= (1.f / 16.f);
  float var = 0.f;
#pragma unroll
  for (int j = 0; j < 16; ++j) { const float d = h2[j] - mu; var += d * d; }
  var *= (1.f / 16.f);
  const float inv = rsqrtf(var + 1e-5f);
  float hn[16];
#pragma unroll
  for (int j = 0; j < 16; ++j) hn[j] = (h2[j] - mu) * inv * lng[j] + lnb[j];
#pragma unroll
  for (int j = 0; j < 16; ++j) {
    float s = b3[j];
#pragma unroll
    for (int k = 0; k < 16; ++k) s += hn[k] * W3[j * 16 + k];
    h3[r * 16 + j] = s;
  }
}

// ---------------------------------------------------------------------------
// Kernel C1: latent = mean(h3, axis=0);  hc = relu(latent@cap_w1^T + cap_b1)
// ---------------------------------------------------------------------------
__global__ __launch_bounds__(256) void k_latent_hidden(
    const float* __restrict__ h3, const float* __restrict__ cw1,
    const float* __restrict__ cb1, float* __restrict__ hc) {
  __shared__ float red[256 * 16];
  __shared__ float lat[16];
  const int t = threadIdx.x;
  float acc[16];
#pragma unroll
  for (int j = 0; j < 16; ++j) acc[j] = 0.f;
  for (int r = t; r < NB; r += 256)
#pragma unroll
    for (int j = 0; j < 16; ++j) acc[j] += h3[r * 16 + j];
#pragma unroll
  for (int j = 0; j < 16; ++j) red[t * 16 + j] = acc[j];
  __syncthreads();
  for (int s = 128; s > 0; s >>= 1) {
    if (t < s)
#pragma unroll
      for (int j = 0; j < 16; ++j) red[t * 16 + j] += red[(t + s) * 16 + j];
    __syncthreads();
  }
  if (t < 16) lat[t] = red[t] * (1.f / (float)NB);
  __syncthreads();
  if (t < 16) {
    float s = cb1[t];
#pragma unroll
    for (int k = 0; k < 16; ++k) s += lat[k] * cw1[t * 16 + k];
    hc[t] = s > 0.f ? s : 0.f;
  }
}

// ---------------------------------------------------------------------------
// Kernel C2: f[mask] = 0.1*sigmoid(hc @ cap_w2[mask-1] + cap_b2[mask-1]); f[0]=0
// ---------------------------------------------------------------------------
__global__ __launch_bounds__(256) void k_increments(
    const float* __restrict__ hc, const float* __restrict__ cw2,
    const float* __restrict__ cb2, float* __restrict__ f) {
  const int m = blockIdx.x * blockDim.x + threadIdx.x;  // 0..4095
  if (m >= 4096) return;
  if (m == 0) { f[0] = 0.f; return; }
  float z = cb2[m - 1];
#pragma unroll
  for (int k = 0; k < 16; ++k) z += hc[k] * cw2[(m - 1) * 16 + k];
  f[m] = 0.1f / (1.f + expf(-z));
}

// ---------------------------------------------------------------------------
// Kernel D: subset-sum (zeta) transform over 4096 masks in LDS, then normalize
// by f[full]+1e-8. Writes cap_lookup (bitmask indexed, lookup[0]=0) back to ws
// and the caps output (masks 1..4095) to d_out tail.
// Replaces the 4095x4095 inclusion matvec (16.7M MACs) with D*2^D adds.
// ---------------------------------------------------------------------------
__global__ __launch_bounds__(1024) void k_zeta_norm(float* __restrict__ f,
                                                    float* __restrict__ caps_out) {
  __shared__ float sf[4096];
  const int t = threadIdx.x;
#pragma unroll
  for (int q = 0; q < 4; ++q) sf[t + 1024 * q] = f[t + 1024 * q];
  __syncthreads();
  for (int d = 0; d < DF; ++d) {
    const int bit = 1 << d;
#pragma unroll
    for (int q = 0; q < 4; ++q) {
      const int m = t + 1024 * q;
      if (m & bit) sf[m] += sf[m ^ bit];  // reads touch only bit-clear masks: race-free
    }
    __syncthreads();
  }
  const float inv = 1.f / (sf[4095] + 1e-8f);
#pragma unroll
  for (int q = 0; q < 4; ++q) {
    const int m = t + 1024 * q;
    const float c = (m == 0) ? 0.f : sf[m] * inv;
    f[m] = c;                       // cap_lookup by bitmask
    if (m >= 1) caps_out[m - 1] = c;  // caps output, subset i <-> bitmask i+1
  }
}

// ---------------------------------------------------------------------------
// Kernel E: one block per query row. Choquet sims for all 1024 neighbors
// (cap LUT in LDS), fused leave-one-out top-32 + softmax(T=0.5) + label vote.
// Sims row never leaves LDS -> zero HBM traffic for the [N,N] matrix.
// ---------------------------------------------------------------------------
__global__ __launch_bounds__(256) void k_knn(const float* __restrict__ X,
                                             const int* __restrict__ y,
                                             const float* __restrict__ capl,
                                             float* __restrict__ votes) {
  __shared__ float scap[4096];   // 16 KB capacity LUT
  __shared__ float ssims[NB];    // 4 KB sims row
  __shared__ float rv[256];
  __shared__ int   ri[256];
  __shared__ float topv[KNB];
  __shared__ int   topi[KNB];
  __shared__ float sexp[KNB];
  __shared__ float svotes[NCLS];
  __shared__ float ssum;
  const int tid = threadIdx.x;
  const int i   = blockIdx.x;

  for (int p = tid; p < 4096; p += 256) scap[p] = capl[p];
  if (tid < NCLS) svotes[tid] = 0.f;
  __syncthreads();

  float xi[DF];
#pragma unroll
  for (int d = 0; d < DF; ++d) xi[d] = X[i * DF + d];

  for (int jj = tid; jj < NB; jj += 256) {
    float v[DF];
    int   bm[DF];
#pragma unroll
    for (int d = 0; d < DF; ++d) {
      v[d]  = expf(-fabsf(xi[d] - X[jj * DF + d]));
      bm[d] = 1 << d;
    }
    // Stable odd-even transposition sort, ascending; fully unrolled -> registers.
#define CSW(ii, jjj)                                                       \
    do {                                                                   \
      if (v[ii] > v[jjj]) {                                                \
        float tv = v[ii]; v[ii] = v[jjj]; v[jjj] = tv;                     \
        int tb = bm[ii]; bm[ii] = bm[jjj]; bm[jjj] = tb;                   \
      }                                                                    \
    } while (0)
#pragma unroll
    for (int r = 0; r < DF; ++r) {
#pragma unroll
      for (int p = 0; p + 1 < DF; ++p) {
        if ((p & 1) == (r & 1)) CSW(p, p + 1);
      }
    }
#undef CSW
    // tail[t] = OR of feature bits at sorted positions t..D-1
    int suf[DF + 1];
    suf[DF] = 0;
#pragma unroll
    for (int t2 = DF - 1; t2 >= 0; --t2) suf[t2] = suf[t2 + 1] | bm[t2];
    float prev = 0.f, acc = 0.f;
#pragma unroll
    for (int t2 = 0; t2 < DF; ++t2) {
      acc += (v[t2] - prev) * scap[suf[t2]];
      prev = v[t2];
    }
    if (jj == i) acc -= 1e9f;  // leave-one-out
    ssims[jj] = acc;
  }
  __syncthreads();

  // 32 rounds of argmax; ties -> lowest index (matches jax.lax.top_k)
  for (int it = 0; it < KNB; ++it) {
    float bv = -3.0e38f;
    int   bi = 0x7fffffff;
    for (int jj = tid; jj < NB; jj += 256) {
      const float s = ssims[jj];
      if (s > bv) { bv = s; bi = jj; }
    }
    rv[tid] = bv; ri[tid] = bi;
    __syncthreads();
    for (int st = 128; st > 0; st >>= 1) {
      if (tid < st) {
        if (rv[tid + st] > rv[tid] ||
            (rv[tid + st] == rv[tid] && ri[tid + st] < ri[tid])) {
          rv[tid] = rv[tid + st];
          ri[tid] = ri[tid + st];
        }
      }
      __syncthreads();
    }
    if (tid == 0) {
      topv[it] = rv[0];
      topi[it] = ri[0];
      ssims[ri[0]] = -3.0e38f;
    }
    __syncthreads();
  }

  // softmax(top_sims / 0.5) and class vote
  if (tid < KNB) sexp[tid] = expf((topv[tid] - topv[0]) * 2.0f);
  __syncthreads();
  if (tid == 0) {
    float s = 0.f;
    for (int k = 0; k < KNB; ++k) s += sexp[k];
    ssum = s;
  }
  __syncthreads();
  if (tid < KNB) atomicAdd(&svotes[y[topi[tid]]], sexp[tid] / ssum);
  __syncthreads();
  if (tid < NCLS) votes[i * NCLS + tid] = svotes[tid];
}

// ---------------------------------------------------------------------------
extern "C" void kernel_launch(void* const* d_in, const int* in_sizes, int n_in,
                              void* d_out, int out_size, void* d_ws, size_t ws_size,
                              hipStream_t stream) {
  (void)in_sizes; (void)n_in; (void)out_size; (void)ws_size;
  const float* X   = (const float*)d_in[0];
  const int*   y   = (const int*)d_in[1];
  const float* w1  = (const float*)d_in[2];
  const float* b1  = (const float*)d_in[3];
  const float* w2  = (const float*)d_in[4];
  const float* b2  = (const float*)d_in[5];
  const float* lng = (const float*)d_in[6];
  const float* lnb = (const float*)d_in[7];
  const float* w3  = (const float*)d_in[8];
  const float* b3  = (const float*)d_in[9];
  const float* cw1 = (const float*)d_in[10];
  const float* cb1 = (const float*)d_in[11];
  const float* cw2 = (const float*)d_in[12];
  const float* cb2 = (const float*)d_in[13];

  float* out      = (float*)d_out;
  float* votes    = out;                 // [1024,10]
  float* caps_out = out + NB * NCLS;     // [4095]

  char*  ws = (char*)d_ws;
  float* h1 = (float*)(ws);              // 1024*32 f32 = 128 KB
  float* h3 = (float*)(ws + 131072);     // 1024*16 f32 =  64 KB
  float* hc = (float*)(ws + 196608);     // 16 f32
  float* f  = (float*)(ws + 196672);     // 4096 f32 (increments -> cap_lookup)

  k_gemm1_wmma  <<<128, 32,  0, stream>>>(X, w1, b1, h1);
  k_encoder_rest<<<4,   256, 0, stream>>>(h1, w2, b2, lng, lnb, w3, b3, h3);
  k_latent_hidden<<<1,  256, 0, stream>>>(h3, cw1, cb1, hc);
  k_increments  <<<16,  256, 0, stream>>>(hc, cw2, cb2, f);
  k_zeta_norm   <<<1,  1024, 0, stream>>>(f, caps_out);
  k_knn         <<<NB,  256, 0, stream>>>(X, y, f, votes);
}